// position_encode_27779848471294
// MI455X (gfx1250) — compile-verified
//
#include <hip/hip_runtime.h>

typedef __attribute__((ext_vector_type(2))) float v2f;
typedef __attribute__((ext_vector_type(8))) float v8f;

#define ACTION_SIZE 1024
#define SLOTS       5
#define OUT_DIM     16
#define FAN_IN      (SLOTS * ACTION_SIZE)   // 5120
#define BATCH       16384

// ---------------------------------------------------------------------------
// Kernel 1: transpose W [16, 5120] (row-major) -> Wt [5120, 16] in workspace.
// One thread per input column c: the o-loop loads are lane-coalesced (c is
// contiguous across lanes for each fixed o), and each thread writes one
// contiguous 64B line as 4x float4. 327 KB total -> negligible.
// ---------------------------------------------------------------------------
__global__ __launch_bounds__(256) void transpose_w_kernel(const float* __restrict__ W,
                                                          float* __restrict__ Wt) {
    int c = blockIdx.x * blockDim.x + threadIdx.x;
    if (c >= FAN_IN) return;
    float v[OUT_DIM];
#pragma unroll
    for (int o = 0; o < OUT_DIM; ++o) v[o] = W[o * FAN_IN + c];
    float4* dst = (float4*)(Wt + c * OUT_DIM);
#pragma unroll
    for (int q = 0; q < 4; ++q)
        dst[q] = make_float4(v[4 * q + 0], v[4 * q + 1], v[4 * q + 2], v[4 * q + 3]);
}

// ---------------------------------------------------------------------------
// Kernel 2: one wave32 per 16-row tile.
//   out_tile(16x16) = relu( A'(16x80) x B'(80x16) + bias )
//     B' row j  = Wt[ loc[r0*5 + j] ][0..15]          (gathered column, j<80)
//     A'[r][j]  = 1 iff j in [5r, 5r+5)               (constant selection)
// 20x V_WMMA_F32_16X16X4_F32, bias preloaded into C, ReLU + coalesced stores.
// ---------------------------------------------------------------------------
__global__ __launch_bounds__(256) void onehot_linear_wmma_kernel(
        const int* __restrict__ loc, const float* __restrict__ Wt,
        const float* __restrict__ bias, float* __restrict__ out) {
    const int  lane = threadIdx.x & 31;
    const int  wave = threadIdx.x >> 5;
    const int  tile = blockIdx.x * 8 + wave;     // 0..1023
    const int  r0   = tile * 16;                 // first batch row of tile
    const int  n    = lane & 15;                 // N (output) index of this lane
    const bool hi   = lane >= 16;

    // C = bias broadcast along N: C[M][N] = bias[N] for every M.
    const float bv = bias[n];
    v8f acc;
#pragma unroll
    for (int i = 0; i < 8; ++i) acc[i] = bv;

    // A/B 32-bit register layout: VGPR0 holds K-local {0,2}, VGPR1 {1,3}
    // (low lanes get 0/1, high lanes 2/3).
    const int kofs0 = hi ? 2 : 0;
    const int kofs1 = hi ? 3 : 1;
    const int m5    = SLOTS * n;                 // A row M == n for this lane

    const int* locTile = loc + r0 * SLOTS;       // 80 indices for this tile

#pragma unroll 4
    for (int k = 0; k < 20; ++k) {               // 20 * K=4 == 80, exact
        const int j0 = 4 * k + kofs0;
        const int j1 = 4 * k + kofs1;

        // Constant block-diagonal selection matrix A'.
        const float a0 = (j0 >= m5 && j0 < m5 + SLOTS) ? 1.0f : 0.0f;
        const float a1 = (j1 >= m5 && j1 < m5 + SLOTS) ? 1.0f : 0.0f;

        // Gather B' rows: lanes 0-15 read 64B contiguous per column (L2-hot).
        const int c0 = locTile[j0];
        const int c1 = locTile[j1];
        const float b0 = Wt[c0 * OUT_DIM + n];
        const float b1 = Wt[c1 * OUT_DIM + n];

        v2f A = {a0, a1};
        v2f B = {b0, b1};
        acc = __builtin_amdgcn_wmma_f32_16x16x4_f32(
                  /*neg_a=*/false, A, /*neg_b=*/false, B,
                  /*c_mod=*/(short)0, acc, /*reuse_a=*/false, /*reuse_b=*/false);
    }

    // D layout: VGPR v -> M = v (lanes 0-15) or v+8 (lanes 16-31), N = lane%16.
    // Each VGPR store covers two full contiguous 64B output rows.
#pragma unroll
    for (int v = 0; v < 8; ++v) {
        const int M = hi ? (v + 8) : v;
        out[(r0 + M) * OUT_DIM + n] = fmaxf(acc[v], 0.0f);
    }
}

extern "C" void kernel_launch(void* const* d_in, const int* in_sizes, int n_in,
                              void* d_out, int out_size, void* d_ws, size_t ws_size,
                              hipStream_t stream) {
    const int*   loc  = (const int*)d_in[0];     // [16384, 5]
    const float* W    = (const float*)d_in[1];   // [16, 5120]
    const float* bias = (const float*)d_in[2];   // [16]
    float*       out  = (float*)d_out;           // [16384, 16]
    float*       Wt   = (float*)d_ws;            // [5120, 16] scratch (327,680 B)

    transpose_w_kernel<<<(FAN_IN + 255) / 256, 256, 0, stream>>>(W, Wt);

    // 16384 rows / 16 rows-per-tile = 1024 tiles; 8 waves (tiles) per block.
    onehot_linear_wmma_kernel<<<(BATCH / 16) / 8, 256, 0, stream>>>(loc, Wt, bias, out);
}